// PointPillarsScatter_16355235463869
// MI455X (gfx1250) — compile-verified
//
#include <hip/hip_runtime.h>
#include <hip/hip_bf16.h>
#include <stdint.h>

// ---------------------------------------------------------------------------
// PointPillarsScatter for MI455X (gfx1250).
// Memory-bound: 219.4 MB zero-fill + 16.4 MB scatter -> ~11 us at 23.3 TB/s.
// No FLOPs -> no WMMA; the CDNA5 feature that fits is the Tensor Data Mover:
// each workgroup issues ONE iterated tensor_store_from_lds descriptor that
// DMA-streams up to 256 x 32KB zero tiles from LDS to HBM (TENSORcnt).
// ---------------------------------------------------------------------------

#define TILE_ELEMS 8192          // 32 KB tile of fp32 zeros staged in LDS
#define TILE_BYTES (TILE_ELEMS * 4)

typedef unsigned int u32x4 __attribute__((ext_vector_type(4)));
typedef int          i32x8 __attribute__((ext_vector_type(8)));
typedef int          i32x4 __attribute__((ext_vector_type(4)));

// ---------------------------------------------------------------------------
// Kernel 1: zero the canvas via TDM stores from an LDS zero tile.
// tiles/per/tail bounds are host-computed (no device-side 64-bit division).
// ---------------------------------------------------------------------------
__global__ void pp_fill_zero_tdm(float* __restrict__ out,
                                 long long n,          // total elements
                                 int tiles,            // n / TILE_ELEMS
                                 int per) {            // ceil(tiles / gridDim.x)
    __shared__ float4 zbuf[TILE_ELEMS / 4];            // 32 KB of zeros

    // Cooperatively zero the LDS staging tile (b128 DS stores).
    for (int i = threadIdx.x; i < TILE_ELEMS / 4; i += blockDim.x)
        zbuf[i] = make_float4(0.f, 0.f, 0.f, 0.f);
    __syncthreads();  // DS writes visible before the TDM reads LDS

    int t0 = blockIdx.x * per;
    int t1 = t0 + per; if (t1 > tiles) t1 = tiles;

#if defined(__gfx1250__) && __has_builtin(__builtin_amdgcn_tensor_store_from_lds)
    if (threadIdx.x == 0 && t0 < t1) {
        // addrspace(3) offset: flat LDS addresses carry the LDS byte offset
        // in addr[31:0] (ISA 10.2 aperture mapping), so truncation is exact.
        unsigned lds_addr = (unsigned)(uintptr_t)(&zbuf[0]);

        for (int t = t0; t < t1; t += 256) {
            int left  = t1 - t;
            int iters = left < 256 ? left : 256;       // tiles in this DMA op
            unsigned long long ga =
                (unsigned long long)(uintptr_t)out + (unsigned long long)t * TILE_BYTES;

            // ---- D# group 0 (ISA 8.3): count=1 | lds_addr | global_addr | type=2
            u32x4 g0 = {
                1u,                                             // count=1 (valid)
                lds_addr,                                       // LDS byte address
                (unsigned)(ga & 0xFFFFFFFFull),                 // global_addr[31:0]
                (unsigned)((ga >> 32) & 0x01FFFFFFull) | 0x80000000u // [56:32] | type=2
            };
            // ---- D# group 1 (ISA 8.4): data_size=4B, iterate_enable,
            //      tensor 8192x1, tile 8192x1, dim0_stride=8192
            i32x8 g1 = {
                (2 << 16) | (1 << 19),     // data_size=2(4B), iterate_enable=1
                (int)(TILE_ELEMS << 16),   // [31:16] tensor_dim0[15:0]
                (int)(1 << 16),            // [31:16] tensor_dim1[15:0] = 1
                (int)(TILE_ELEMS << 16),   // [31:16] tile_dim0 = 8192
                1,                         // [15:0]  tile_dim1 = 1; tile_dim2 = 0
                TILE_ELEMS,                // tensor_dim0_stride[31:0]
                0,                         // dim0_stride[47:32]=0; dim1_stride[15:0]=0
                0                          // tensor_dim1_stride[47:16]=0
            };
            // ---- D# group 2 (ISA 8.5, iterate mode): lds_incr=0 (reuse zeros),
            //      global_incr=8192 elems (=32KB), iterate_count=iters-1
            i32x4 g2 = { 0, 0, TILE_ELEMS, (int)((iters - 1) << 16) };
            i32x4 g3 = { 0, 0, 0, 0 };
            // Trailing group required by the 6-arg builtin on this toolchain;
            // all-reserved/zero for our simple 2D iterated tile.
            i32x8 gx = { 0, 0, 0, 0, 0, 0, 0, 0 };

            __builtin_amdgcn_tensor_store_from_lds(g0, g1, g2, g3, gx, 0);
        }
#if __has_builtin(__builtin_amdgcn_s_wait_tensorcnt)
        __builtin_amdgcn_s_wait_tensorcnt(0);
#endif
    }
#else
    // Fallback: plain b128 grid-stride memset over the tile-covered region.
    {
        float4* o4 = (float4*)out;
        long long n4 = ((long long)tiles * TILE_ELEMS) / 4;
        long long gtid    = (long long)blockIdx.x * blockDim.x + threadIdx.x;
        long long gstride = (long long)gridDim.x * blockDim.x;
        for (long long i = gtid; i < n4; i += gstride)
            o4[i] = make_float4(0.f, 0.f, 0.f, 0.f);
    }
#endif

    // Tail (n not a multiple of TILE_ELEMS) — empty for this problem size.
    {
        long long start   = (long long)tiles * TILE_ELEMS;
        long long gtid    = (long long)blockIdx.x * blockDim.x + threadIdx.x;
        long long gstride = (long long)gridDim.x * blockDim.x;
        for (long long k = start + gtid; k < n; k += gstride)
            out[k] = 0.f;
    }
}

// ---------------------------------------------------------------------------
// Kernel 2: scatter. One thread per (pillar, 4-channel group): coalesced
// float4 feature read, 4 scattered b32 stores at stride `spatial`.
// out[(b*C + c)*spatial + z*nz + y*nx + x] = vf[p, c]   (reference formula)
// ---------------------------------------------------------------------------
__global__ void pp_scatter4(const float4* __restrict__ vf4,
                            const int* __restrict__ coords,
                            const int* __restrict__ nz_p,
                            const int* __restrict__ ny_p,
                            const int* __restrict__ nx_p,
                            float* __restrict__ out,
                            int N, int Cq, int C) {
    long long tid = (long long)blockIdx.x * blockDim.x + threadIdx.x;
    long long total = (long long)N * Cq;
    if (tid >= total) return;

    int p  = (int)(tid / Cq);
    int cq = (int)(tid - (long long)p * Cq);

    float4 v = vf4[tid];                      // element offset = p*C + 4*cq  (coalesced)

    int b = coords[4 * p + 0];
    int z = coords[4 * p + 1];
    int y = coords[4 * p + 2];
    int x = coords[4 * p + 3];

    int nz = *nz_p, ny = *ny_p, nx = *nx_p;   // uniform scalar loads
    long long spatial = (long long)nz * ny * nx;
    long long s = (long long)z * nz + (long long)y * nx + x;   // faithful to reference

    float* dst = out + ((long long)b * C + 4 * cq) * spatial + s;
    dst[0]           = v.x;
    dst[spatial]     = v.y;
    dst[2 * spatial] = v.z;
    dst[3 * spatial] = v.w;
}

// Scalar fallback for C % 4 != 0 (not hit for C=64, kept for generality).
__global__ void pp_scatter1(const float* __restrict__ vf,
                            const int* __restrict__ coords,
                            const int* __restrict__ nz_p,
                            const int* __restrict__ ny_p,
                            const int* __restrict__ nx_p,
                            float* __restrict__ out,
                            int N, int C) {
    long long tid = (long long)blockIdx.x * blockDim.x + threadIdx.x;
    long long total = (long long)N * C;
    if (tid >= total) return;

    int p = (int)(tid / C);
    int c = (int)(tid - (long long)p * C);

    int b = coords[4 * p + 0];
    int z = coords[4 * p + 1];
    int y = coords[4 * p + 2];
    int x = coords[4 * p + 3];

    int nz = *nz_p, ny = *ny_p, nx = *nx_p;
    long long spatial = (long long)nz * ny * nx;
    long long s = (long long)z * nz + (long long)y * nx + x;

    out[((long long)b * C + c) * spatial + s] = vf[tid];
}

// ---------------------------------------------------------------------------
// Launch: fill (TDM memset) then scatter, sequential on `stream`.
// Inputs: [0]=voxel_features f32, [1]=coords i32[N,4], [2]=batch_size,
//         [3]=nz, [4]=ny, [5]=nx  (scalars as 1-elem int arrays)
// ---------------------------------------------------------------------------
extern "C" void kernel_launch(void* const* d_in, const int* in_sizes, int n_in,
                              void* d_out, int out_size, void* d_ws, size_t ws_size,
                              hipStream_t stream) {
    (void)n_in; (void)d_ws; (void)ws_size;

    const float* vf     = (const float*)d_in[0];
    const int*   coords = (const int*)d_in[1];
    const int*   nz_p   = (const int*)d_in[3];
    const int*   ny_p   = (const int*)d_in[4];
    const int*   nx_p   = (const int*)d_in[5];
    float*       out    = (float*)d_out;

    int N = in_sizes[1] / 4;                       // pillar count
    int C = (N > 0) ? (in_sizes[0] / N) : 0;       // channels (64)

    // 1) Zero the canvas: 384 WGs x 256 threads; each WG DMA-stores its slice.
    const int fillWG = 384;
    int tiles = (int)((long long)out_size / TILE_ELEMS);   // 6696 for this shape
    int per   = (tiles + fillWG - 1) / fillWG;             // 18 tiles (576 KB) per WG
    pp_fill_zero_tdm<<<fillWG, 256, 0, stream>>>(out, (long long)out_size, tiles, per);

    // 2) Scatter features into the zeroed canvas.
    if (N > 0 && C > 0) {
        if ((C & 3) == 0) {
            int Cq = C >> 2;
            long long total = (long long)N * Cq;
            int blocks = (int)((total + 255) / 256);
            pp_scatter4<<<blocks, 256, 0, stream>>>((const float4*)vf, coords,
                                                    nz_p, ny_p, nx_p, out, N, Cq, C);
        } else {
            long long total = (long long)N * C;
            int blocks = (int)((total + 255) / 256);
            pp_scatter1<<<blocks, 256, 0, stream>>>(vf, coords,
                                                    nz_p, ny_p, nx_p, out, N, C);
        }
    }
}